// InnerPropagation_38646115729710
// MI455X (gfx1250) — compile-verified
//
#include <hip/hip_runtime.h>
#include <hip/hip_bf16.h>

#define D 256                     // D_IN == D_OUT
#define LDS_STRIDE (D + 1)        // +1 float pad: row r starts at bank (r%64), kills conflicts
#define WAVES_PER_BLOCK 8
#define ROWS_PER_BLOCK (WAVES_PER_BLOCK * 16)   // 128 rows per workgroup
#define SMEM_FLOATS ((ROWS_PER_BLOCK + 16) * LDS_STRIDE)
#define SMEM_BYTES (SMEM_FLOATS * sizeof(float))

typedef float v2f __attribute__((ext_vector_type(2)));
typedef float v8f __attribute__((ext_vector_type(8)));

// ---------------------------------------------------------------------------
// Utility: zero fill
// ---------------------------------------------------------------------------
__global__ void zero_f32_kernel(float* __restrict__ p, size_t n) {
    size_t i = (size_t)blockIdx.x * blockDim.x + threadIdx.x;
    if (i < n) p[i] = 0.0f;
}
__global__ void zero_u32_kernel(unsigned* __restrict__ p, size_t n) {
    size_t i = (size_t)blockIdx.x * blockDim.x + threadIdx.x;
    if (i < n) p[i] = 0u;
}

// ---------------------------------------------------------------------------
// Kernel 1: out = A @ W^T + bias   (A:[rows,256], W:[256,256] row-major)
// 8 waves/block, 128 A-rows staged in LDS once, W j-tiles staged cooperatively
// (coalesced), fp32 WMMA frags read from padded LDS.
// ---------------------------------------------------------------------------
__global__ __launch_bounds__(256) void wmma_fc_kernel(
        const float* __restrict__ A, const float* __restrict__ W,
        const float* __restrict__ bias, float* __restrict__ out, int rows) {
    extern __shared__ float smem[];
    float* lds_a = smem;                                   // 128 x 257
    float* lds_w = smem + ROWS_PER_BLOCK * LDS_STRIDE;     // 16 x 257

    const int tid  = threadIdx.x;          // 0..255
    const int wave = tid >> 5;
    const int lane = tid & 31;
    const int n0   = blockIdx.x * ROWS_PER_BLOCK;

    // stage the 128x256 A tile (coalesced; pad rows beyond `rows` with zeros)
    for (int t = tid; t < ROWS_PER_BLOCK * D; t += 256) {
        int r = t >> 8, k = t & (D - 1);
        int gr = n0 + r;
        lds_a[r * LDS_STRIDE + k] = (gr < rows) ? A[(size_t)gr * D + k] : 0.0f;
    }

    const int lm   = lane & 15;
    const int koff = (lane >> 4) * 2;      // K=0/1 (lanes 0-15), K=2/3 (16-31)
    const float* arow = lds_a + (wave * 16 + lm) * LDS_STRIDE;

    for (int jt = 0; jt < D / 16; ++jt) {
        const int j0 = jt * 16;
        __syncthreads();                                   // guards A stage + prev W
        for (int t = tid; t < 16 * D; t += 256) {
            int r = t >> 8, k = t & (D - 1);
            lds_w[r * LDS_STRIDE + k] = W[(size_t)(j0 + r) * D + k];
        }
        __syncthreads();

        const float* wrow = lds_w + lm * LDS_STRIDE;
        v8f c = {};
        for (int k0 = 0; k0 < D; k0 += 4) {
            v2f a, b;
            a.x = arow[k0 + koff];
            a.y = arow[k0 + koff + 1];
            b.x = wrow[k0 + koff];
            b.y = wrow[k0 + koff + 1];
            c = __builtin_amdgcn_wmma_f32_16x16x4_f32(false, a, false, b,
                                                      (short)0, c, false, false);
        }

        // C layout: lanes 0-15 hold M=r, lanes 16-31 hold M=r+8; N=lm
        const int rbase = n0 + wave * 16 + ((lane >> 4) << 3);
        const int col   = j0 + lm;
        const float bv  = bias[col];
#pragma unroll
        for (int r = 0; r < 8; ++r) {
            int row = rbase + r;
            if (row < rows) out[(size_t)row * D + col] = c[r] + bv;
        }
    }
}

// ---------------------------------------------------------------------------
// Kernel 2: s[n] = sum_j leaky_relu(x_n . we_w[j] + b[j]) * c_e[j]
// Same blocking; epilogue fuses leaky_relu, c_e weighting and lane reduction.
// ---------------------------------------------------------------------------
__global__ __launch_bounds__(256) void wmma_logits_kernel(
        const float* __restrict__ X, const float* __restrict__ W,
        const float* __restrict__ bias, const float* __restrict__ ce,
        float* __restrict__ s, int n_nodes) {
    extern __shared__ float smem[];
    float* lds_a = smem;
    float* lds_w = smem + ROWS_PER_BLOCK * LDS_STRIDE;

    const int tid  = threadIdx.x;
    const int wave = tid >> 5;
    const int lane = tid & 31;
    const int n0   = blockIdx.x * ROWS_PER_BLOCK;

    for (int t = tid; t < ROWS_PER_BLOCK * D; t += 256) {
        int r = t >> 8, k = t & (D - 1);
        int gr = n0 + r;
        lds_a[r * LDS_STRIDE + k] = (gr < n_nodes) ? X[(size_t)gr * D + k] : 0.0f;
    }

    const int lm   = lane & 15;
    const int koff = (lane >> 4) * 2;
    const float* arow = lds_a + (wave * 16 + lm) * LDS_STRIDE;

    float psum[8] = {0.f, 0.f, 0.f, 0.f, 0.f, 0.f, 0.f, 0.f};

    for (int jt = 0; jt < D / 16; ++jt) {
        const int j0 = jt * 16;
        __syncthreads();
        for (int t = tid; t < 16 * D; t += 256) {
            int r = t >> 8, k = t & (D - 1);
            lds_w[r * LDS_STRIDE + k] = W[(size_t)(j0 + r) * D + k];
        }
        __syncthreads();

        const float* wrow = lds_w + lm * LDS_STRIDE;
        v8f c = {};
        for (int k0 = 0; k0 < D; k0 += 4) {
            v2f a, b;
            a.x = arow[k0 + koff];
            a.y = arow[k0 + koff + 1];
            b.x = wrow[k0 + koff];
            b.y = wrow[k0 + koff + 1];
            c = __builtin_amdgcn_wmma_f32_16x16x4_f32(false, a, false, b,
                                                      (short)0, c, false, false);
        }

        const int col   = j0 + lm;
        const float bv  = bias[col];
        const float cv  = ce[col];
#pragma unroll
        for (int r = 0; r < 8; ++r) {
            float v = c[r] + bv;
            v = (v > 0.0f) ? v : 0.01f * v;        // leaky_relu(0.01)
            psum[r] += v * cv;
        }
    }

    // reduce across the 16 lanes of each half-wave (halves hold M and M+8)
#pragma unroll
    for (int r = 0; r < 8; ++r) {
        float v = psum[r];
        v += __shfl_xor(v, 8, 16);
        v += __shfl_xor(v, 4, 16);
        v += __shfl_xor(v, 2, 16);
        v += __shfl_xor(v, 1, 16);
        psum[r] = v;
    }
    const int rbase = n0 + wave * 16 + ((lane >> 4) << 3);
    if (lm == 0) {
#pragma unroll
        for (int r = 0; r < 8; ++r) {
            int row = rbase + r;
            if (row < n_nodes) s[row] = psum[r];
        }
    }
}

// ---------------------------------------------------------------------------
// Kernel 3: incidence structure. bitmap mode: distinct-edge bitmask per node;
// fallback mode: pair multiplicity counts.
// ---------------------------------------------------------------------------
__global__ void incidence_kernel(const int* __restrict__ nid,
                                 const int* __restrict__ eid,
                                 unsigned* __restrict__ ws, int m,
                                 int wpn, int use_bitmap) {
    int i = blockIdx.x * blockDim.x + threadIdx.x;
    if (i >= m) return;
    int n = nid[i];
    if (use_bitmap) {
        int e = eid[i];
        atomicOr(&ws[(size_t)n * wpn + (e >> 5)], 1u << (e & 31));
    } else {
        atomicAdd(&ws[n], 1u);
    }
}

// ---------------------------------------------------------------------------
// Kernel 4: closed-form softmax coefficient per node.
// a_n = exp(s-m) / (k*exp(s-m) + (E-k)*exp(-m)),  m = max(s,0);  a_n=0 if k==0
// ---------------------------------------------------------------------------
__global__ void attn_coef_kernel(const float* __restrict__ s,
                                 const unsigned* __restrict__ ws,
                                 float* __restrict__ a, int n_nodes,
                                 int e_total, int wpn, int use_bitmap) {
    int n = blockIdx.x * blockDim.x + threadIdx.x;
    if (n >= n_nodes) return;
    int k;
    if (use_bitmap) {
        const unsigned* w = ws + (size_t)n * wpn;
        int c = 0;
        for (int i = 0; i < wpn; ++i) c += __popc(w[i]);
        k = c;
    } else {
        k = (int)ws[n];
        if (k > e_total) k = e_total;
    }
    float res = 0.0f;
    if (k > 0) {
        float sv = s[n];
        float mx = fmaxf(sv, 0.0f);
        float es = __expf(sv - mx);
        float e0 = __expf(-mx);
        res = es / ((float)k * es + (float)(e_total - k) * e0);
    }
    a[n] = res;
}

// ---------------------------------------------------------------------------
// Kernel 5: float4 atomic scatter-add of raw he_t rows into out (unscaled).
// One thread per (pair, 16-byte chunk): M * 64 threads. out + he_t are
// L2-resident (51 MB + 2 MB << 192 MB), so atomics never touch HBM.
// ---------------------------------------------------------------------------
__global__ void scatter_kernel(const int* __restrict__ nid,
                               const int* __restrict__ eid,
                               const float* __restrict__ he_t,
                               float* __restrict__ out, int m) {
    long long t = (long long)blockIdx.x * blockDim.x + threadIdx.x;
    long long total = (long long)m * (D / 4);
    if (t >= total) return;
    int pair = (int)(t >> 6);
    int c    = (int)(t & 63);
    int n = nid[pair];
    int e = eid[pair];
    const float4 v = ((const float4*)he_t)[(size_t)e * (D / 4) + c];
    float* dst = out + (size_t)n * D + c * 4;
    atomicAdd(dst + 0, v.x);
    atomicAdd(dst + 1, v.y);
    atomicAdd(dst + 2, v.z);
    atomicAdd(dst + 3, v.w);
}

// ---------------------------------------------------------------------------
// Kernel 6: out = relu(a[n] * acc)
// ---------------------------------------------------------------------------
__global__ void finalize_kernel(float* __restrict__ out,
                                const float* __restrict__ a, size_t total) {
    size_t i = (size_t)blockIdx.x * blockDim.x + threadIdx.x;
    if (i >= total) return;
    float v = out[i] * a[i >> 8];    // D == 256
    out[i] = fmaxf(v, 0.0f);
}

// ---------------------------------------------------------------------------
// Host launcher
// ---------------------------------------------------------------------------
extern "C" void kernel_launch(void* const* d_in, const int* in_sizes, int n_in,
                              void* d_out, int out_size, void* d_ws, size_t ws_size,
                              hipStream_t stream) {
    const float* node_emb = (const float*)d_in[0];
    const float* he_emb   = (const float*)d_in[1];
    const int*   node_ids = (const int*)d_in[2];
    const int*   edge_ids = (const int*)d_in[3];
    const float* fc_w     = (const float*)d_in[4];
    const float* fc_b     = (const float*)d_in[5];
    const float* we_w     = (const float*)d_in[6];
    const float* we_b     = (const float*)d_in[7];
    const float* c_e      = (const float*)d_in[8];
    float* out = (float*)d_out;

    const int N = in_sizes[0] / D;
    const int E = in_sizes[1] / D;
    const int M = in_sizes[2];
    const int wpn = (E + 31) / 32;               // bitmap words per node

    // workspace layout
    char* ws = (char*)d_ws;
    size_t he_off  = 0;
    size_t s_off   = he_off + (size_t)E * D * sizeof(float);
    size_t a_off   = s_off + (size_t)N * sizeof(float);
    size_t inc_off = a_off + (size_t)N * sizeof(float);
    size_t bitmap_bytes = (size_t)N * wpn * sizeof(unsigned);

    int use_bitmap = (ws_size >= inc_off + bitmap_bytes) ? 1 : 0;

    float*    he_t  = (float*)(ws + he_off);
    float*    s_buf = (float*)(ws + s_off);
    float*    a_buf = (float*)(ws + a_off);
    unsigned* inc   = (unsigned*)(ws + inc_off);

    // 1) zero output accumulator and incidence workspace
    {
        size_t n = (size_t)N * D;
        zero_f32_kernel<<<(unsigned)((n + 255) / 256), 256, 0, stream>>>(out, n);
        size_t w = use_bitmap ? (size_t)N * wpn : (size_t)N;
        zero_u32_kernel<<<(unsigned)((w + 255) / 256), 256, 0, stream>>>(inc, w);
    }

    // 2) he_t = he_emb @ fc_w^T + fc_b   (fp32 WMMA, LDS-blocked)
    {
        unsigned blocks = (unsigned)((E + ROWS_PER_BLOCK - 1) / ROWS_PER_BLOCK);
        wmma_fc_kernel<<<blocks, 256, SMEM_BYTES, stream>>>(he_emb, fc_w, fc_b,
                                                            he_t, E);
    }

    // 3) s[n] = leaky_relu(x_n @ we_w^T + we_b) . c_e   (fp32 WMMA, LDS-blocked)
    {
        unsigned blocks = (unsigned)((N + ROWS_PER_BLOCK - 1) / ROWS_PER_BLOCK);
        wmma_logits_kernel<<<blocks, 256, SMEM_BYTES, stream>>>(node_emb, we_w,
                                                                we_b, c_e,
                                                                s_buf, N);
    }

    // 4) incidence bitmap / counts
    incidence_kernel<<<(M + 255) / 256, 256, 0, stream>>>(node_ids, edge_ids,
                                                          inc, M, wpn, use_bitmap);

    // 5) per-node attention coefficient
    attn_coef_kernel<<<(N + 255) / 256, 256, 0, stream>>>(s_buf, inc, a_buf, N,
                                                          E, wpn, use_bitmap);

    // 6) scatter-add he_t rows (L2-resident atomics)
    {
        long long total = (long long)M * (D / 4);
        unsigned blocks = (unsigned)((total + 255) / 256);
        scatter_kernel<<<blocks, 256, 0, stream>>>(node_ids, edge_ids, he_t,
                                                   out, M);
    }

    // 7) out = relu(a_n * acc)
    {
        size_t total = (size_t)N * D;
        finalize_kernel<<<(unsigned)((total + 255) / 256), 256, 0, stream>>>(
            out, a_buf, total);
    }
}